// LSTMMultistep_15238543966770
// MI455X (gfx1250) — compile-verified
//
#include <hip/hip_runtime.h>
#include <stdint.h>

// Problem constants (from reference): B=256, T=2048, L=4, H=64, P=1
#define BB 256
#define TT 2048
#define LL 4
#define HH 64

typedef uint32_t u32x4 __attribute__((ext_vector_type(4)));
typedef int32_t  i32x8 __attribute__((ext_vector_type(8)));
typedef int32_t  i32x4 __attribute__((ext_vector_type(4)));

#if __has_builtin(__builtin_amdgcn_exp2f)
__device__ __forceinline__ float fexp(float x) {
  return __builtin_amdgcn_exp2f(x * 1.44269504088896340736f); // e^x = 2^(x*log2e)
}
#else
__device__ __forceinline__ float fexp(float x) { return __expf(x); }
#endif

#if __has_builtin(__builtin_amdgcn_rcpf)
__device__ __forceinline__ float frcp(float x) { return __builtin_amdgcn_rcpf(x); }
#else
__device__ __forceinline__ float frcp(float x) { return 1.0f / x; }
#endif

#if __has_builtin(__builtin_amdgcn_tanhf)
__device__ __forceinline__ float tanhf_(float x) {
  return __builtin_amdgcn_tanhf(x);                       // v_tanh_f32
}
__device__ __forceinline__ float sigmoidf_(float x) {
  return fmaf(0.5f, __builtin_amdgcn_tanhf(0.5f * x), 0.5f); // 1 trans + 2 VALU
}
#else
__device__ __forceinline__ float sigmoidf_(float x) {
  return frcp(1.0f + fexp(-x));                           // v_exp + v_add + v_rcp
}
__device__ __forceinline__ float tanhf_(float x) {
  return 2.0f * sigmoidf_(2.0f * x) - 1.0f;
}
#endif

// ---- wave32 xor-butterfly stages as pure VALU (DPP row_xmask + permlanex16) ----
template <int CTRL>
__device__ __forceinline__ float xmask_add(float x) {
#if __has_builtin(__builtin_amdgcn_update_dpp)
  int v = __builtin_bit_cast(int, x);
  int s = __builtin_amdgcn_update_dpp(0, v, CTRL, 0xf, 0xf, true); // row_xmask
  return x + __builtin_bit_cast(float, s);
#else
  return x + __shfl_xor(x, CTRL - 0x160, 32);
#endif
}

__device__ __forceinline__ float xor16_add(float x) {
#if __has_builtin(__builtin_amdgcn_permlanex16)
  int v = __builtin_bit_cast(int, x);
  int s = __builtin_amdgcn_permlanex16(v, v, 0x76543210, 0xfedcba98, false, false);
  return x + __builtin_bit_cast(float, s);
#else
  return x + __shfl_xor(x, 16, 32);
#endif
}

__device__ __forceinline__ float readlane_f(float x, int lane) {
#if __has_builtin(__builtin_amdgcn_readlane)
  return __builtin_bit_cast(float,
      __builtin_amdgcn_readlane(__builtin_bit_cast(int, x), lane));
#else
  return __shfl(x, lane, 32);
#endif
}

// One wave32 per batch element. Lane j owns hidden units j and j+32.
// LDS holds the T-long scalar sequence for this batch; each layer rewrites it
// in place (block of 32 x-values is read into registers at block start, the
// 32 produced h-values are flushed back at block end).
__global__ __launch_bounds__(32) void lstm_scan_kernel(
    const float* __restrict__ y,      // [B, T, 1]
    const float* __restrict__ W_ih,   // [L, 4H, 1]
    const float* __restrict__ W_hh,   // [L, 4H, 1]
    const float* __restrict__ b_ih,   // [L, 4H]
    const float* __restrict__ b_hh,   // [L, 4H]
    const float* __restrict__ W_hr,   // [L, 1, H]
    const int*   __restrict__ min_seq_len,
    float* __restrict__ out)          // [B, T - msl, 1]
{
  __shared__ __align__(16) float xbuf[TT];

  const int b    = blockIdx.x;
  const int lane = threadIdx.x;

  // ---------------- TDM async load: y[b, :, 0] (TT floats) -> LDS ----------
  const float*   src   = y + (size_t)b * TT;
  const uint64_t gaddr = (uint64_t)(uintptr_t)src;
  const uint32_t laddr = (uint32_t)(uintptr_t)&xbuf[0]; // low 32 bits of flat shared addr = LDS offset

#if __has_builtin(__builtin_amdgcn_tensor_load_to_lds)
  // D# group 0: flags | lds_addr | global_addr | type=2
  u32x4 g0;
  g0[0] = 1u;                                                    // count=1 (valid), user mode
  g0[1] = laddr;                                                 // lds_addr (bytes)
  g0[2] = (uint32_t)gaddr;                                       // global_addr[31:0]
  g0[3] = (uint32_t)((gaddr >> 32) & 0x01FFFFFFu) | (2u << 30);  // global_addr[56:32] | type=2

  // D# group 1: data_size=4B, tensor 2048x1, tile 2048x1, stride0=2048
  i32x8 g1;
  g1[0] = (int)(2u << 16);            // data_size=2 (4 bytes); mask/pad/iterate = 0
  g1[1] = (int)((uint32_t)TT << 16);  // tensor_dim0[15:0] in [31:16]
  g1[2] = (int)(1u << 16);            // tensor_dim0 hi = 0 | tensor_dim1[15:0] = 1
  g1[3] = (int)((uint32_t)TT << 16);  // tensor_dim1 hi = 0 | tile_dim0 = 2048
  g1[4] = 1;                          // tile_dim1 = 1, tile_dim2 = 0
  g1[5] = TT;                         // tensor_dim0_stride lo32
  g1[6] = 0;                          // stride0 hi | tensor_dim1_stride lo
  g1[7] = 0;

  i32x4 gz4 = {0, 0, 0, 0};                      // groups 2/3 unused (<=2D tensor)
  i32x8 gz8 = {0, 0, 0, 0, 0, 0, 0, 0};          // extra descriptor group (clang-23 form)
  __builtin_amdgcn_tensor_load_to_lds(g0, g1, gz4, gz4, gz8, 0);

#if __has_builtin(__builtin_amdgcn_s_wait_tensorcnt)
  __builtin_amdgcn_s_wait_tensorcnt(0);
#else
  asm volatile("s_wait_tensorcnt 0" ::: "memory");
#endif
  asm volatile("" ::: "memory");
#else
  for (int t = lane; t < TT; t += 32) xbuf[t] = src[t];
#endif
  __syncthreads(); // single-wave WG: lowers to S_NOP + memory ordering

  const int msl = min_seq_len[0];

  // ---------------- 4 stacked LSTMP layers, scalar h, 64-wide c ------------
  for (int l = 0; l < LL; ++l) {
    const float* wih = W_ih + l * 4 * HH;
    const float* whh = W_hh + l * 4 * HH;
    const float* bi  = b_ih + l * 4 * HH;
    const float* bh  = b_hh + l * 4 * HH;
    const float* whr = W_hr + l * HH;

    const int u0 = lane, u1 = lane + 32;

    // Per-unit weights into registers (torch gate order i,f,g,o at rows u, 64+u, 128+u, 192+u)
    const float wi_i0 = wih[u0],        wi_f0 = wih[64 + u0], wi_g0 = wih[128 + u0], wi_o0 = wih[192 + u0];
    const float wh_i0 = whh[u0],        wh_f0 = whh[64 + u0], wh_g0 = whh[128 + u0], wh_o0 = whh[192 + u0];
    const float b_i0  = bi[u0] + bh[u0];
    const float b_f0  = bi[64 + u0]  + bh[64 + u0];
    const float b_g0  = bi[128 + u0] + bh[128 + u0];
    const float b_o0  = bi[192 + u0] + bh[192 + u0];
    const float wr0   = whr[u0];

    const float wi_i1 = wih[u1],        wi_f1 = wih[64 + u1], wi_g1 = wih[128 + u1], wi_o1 = wih[192 + u1];
    const float wh_i1 = whh[u1],        wh_f1 = whh[64 + u1], wh_g1 = whh[128 + u1], wh_o1 = whh[192 + u1];
    const float b_i1  = bi[u1] + bh[u1];
    const float b_f1  = bi[64 + u1]  + bh[64 + u1];
    const float b_g1  = bi[128 + u1] + bh[128 + u1];
    const float b_o1  = bi[192 + u1] + bh[192 + u1];
    const float wr1   = whr[u1];

    float h = 0.0f, c0 = 0.0f, c1 = 0.0f;
    float xv    = 0.0f; // 32 preloaded x-values, one per lane, per 32-step block
    float hsave = 0.0f; // lane (t&31) banks step t's h; flushed every 32 steps

    for (int t = 0; t < TT; ++t) {
      const int tm = t & 31;
      if (tm == 0) xv = xbuf[t + lane];              // one coalesced DS read / 32 steps
      const float xt = readlane_f(xv, tm);           // uniform broadcast, no DS

      // unit u0
      float gi0 = fmaf(h, wh_i0, fmaf(xt, wi_i0, b_i0));
      float gf0 = fmaf(h, wh_f0, fmaf(xt, wi_f0, b_f0));
      float gg0 = fmaf(h, wh_g0, fmaf(xt, wi_g0, b_g0));
      float go0 = fmaf(h, wh_o0, fmaf(xt, wi_o0, b_o0));
      float i0  = sigmoidf_(gi0);
      float f0  = sigmoidf_(gf0);
      float g0v = tanhf_(gg0);
      float o0  = sigmoidf_(go0);
      c0 = fmaf(f0, c0, i0 * g0v);
      float s0 = o0 * tanhf_(c0) * wr0;

      // unit u1 (independent chain -> ILP with u0)
      float gi1 = fmaf(h, wh_i1, fmaf(xt, wi_i1, b_i1));
      float gf1 = fmaf(h, wh_f1, fmaf(xt, wi_f1, b_f1));
      float gg1 = fmaf(h, wh_g1, fmaf(xt, wi_g1, b_g1));
      float go1 = fmaf(h, wh_o1, fmaf(xt, wi_o1, b_o1));
      float i1  = sigmoidf_(gi1);
      float f1  = sigmoidf_(gf1);
      float g1v = tanhf_(gg1);
      float o1  = sigmoidf_(go1);
      c1 = fmaf(f1, c1, i1 * g1v);
      float s1 = o1 * tanhf_(c1) * wr1;

      // 64 -> 1 projection: pure-VALU wave32 butterfly (DPP row_xmask + permlanex16);
      // all lanes end with the same value, so h is broadcast for free.
      float p = s0 + s1;
      p = xmask_add<0x161>(p); // xor 1
      p = xmask_add<0x162>(p); // xor 2
      p = xmask_add<0x164>(p); // xor 4
      p = xmask_add<0x168>(p); // xor 8
      p = xor16_add(p);        // xor 16 across row halves
      h = p;

      // bank h in lane (t & 31); flush 32 coalesced DS stores per 32 steps
      hsave = (tm == lane) ? h : hsave;
      if (tm == 31) xbuf[t - 31 + lane] = hsave;     // uniform branch
    }
    __syncthreads(); // intra-wave DS ordering before next layer re-reads xbuf
  }

  // ---------------- drop first msl steps, write [T - msl] floats -----------
  float* ob = out + (size_t)b * (TT - msl);
  for (int t = msl + lane; t < TT; t += 32) ob[t - msl] = xbuf[t];
}

extern "C" void kernel_launch(void* const* d_in, const int* in_sizes, int n_in,
                              void* d_out, int out_size, void* d_ws, size_t ws_size,
                              hipStream_t stream) {
  (void)in_sizes; (void)n_in; (void)d_ws; (void)ws_size; (void)out_size;
  const float* y    = (const float*)d_in[0];
  const float* W_ih = (const float*)d_in[1];
  const float* W_hh = (const float*)d_in[2];
  const float* b_ih = (const float*)d_in[3];
  const float* b_hh = (const float*)d_in[4];
  const float* W_hr = (const float*)d_in[5];
  const int*   msl  = (const int*)d_in[6];
  float* out = (float*)d_out;

  lstm_scan_kernel<<<BB, 32, 0, stream>>>(y, W_ih, W_hh, b_ih, b_hh, W_hr, msl, out);
}